// Torch_Ops_Aten_SegmentReduceModule_66236985639424
// MI455X (gfx1250) — compile-verified
//
#include <hip/hip_runtime.h>
#include <cstdint>

// Segment-sum (aten.segment_reduce 'sum'): out[s] = sum of contiguous rows of data.
// data: [1048576, 64] f32, lengths: [8192] int, out: [8192, 64] f32.
//
// Roofline: 256MB read @ 23.3 TB/s => ~11.5us floor. Pure bandwidth problem;
// no matrix ops wanted. gfx1250 path: async global->LDS streaming (ASYNCcnt).
// One wave32 per segment. Each global_load_async_to_lds_b128 moves 512B =
// TWO rows (lanes 0-15: even row, lanes 16-31: odd row, 16B per lane).
// Depth-8 pair pipeline; float4 accumulate from LDS; shfl_xor(16) combine.

#define N_SEG        8192
#define D            64
#define SCAN_THREADS 256
#define PER_THREAD   (N_SEG / SCAN_THREADS)   // 32

#define WAVES_PER_BLOCK 8
#define PFD   8             // outstanding async pair-ops per wave (wait imm = 7)
#define NBUF  (2 * PFD)     // pair slots per wave; slot rewritten PFD iters after use
#define SLOT_FLOATS 128     // 2 rows * 64 floats = 512B per slot
// LDS: 16 slots * 512B = 8KB/wave, 64KB/block -> 5 blocks (40 waves) per WGP.

// ---------------------------------------------------------------------------
// Kernel 1: exclusive prefix sum of lengths -> offs (int), single block.
// ---------------------------------------------------------------------------
__global__ __launch_bounds__(SCAN_THREADS)
void seg_scan_kernel(const int* __restrict__ lengths, int* __restrict__ offs) {
    __shared__ int partials[SCAN_THREADS];
    const int t    = threadIdx.x;
    const int base = t * PER_THREAD;

    int s = 0;
    for (int j = 0; j < PER_THREAD; ++j) s += lengths[base + j];
    partials[t] = s;
    __syncthreads();

    for (int off = 1; off < SCAN_THREADS; off <<= 1) {
        int v = (t >= off) ? partials[t - off] : 0;
        __syncthreads();
        partials[t] += v;
        __syncthreads();
    }

    int run = (t == 0) ? 0 : partials[t - 1];
    for (int j = 0; j < PER_THREAD; ++j) {
        offs[base + j] = run;
        run += lengths[base + j];
    }
}

// ---------------------------------------------------------------------------
// Kernel 2: one wave32 per segment; async global->LDS pair pipeline (b128).
// ---------------------------------------------------------------------------
__global__ __launch_bounds__(WAVES_PER_BLOCK * 32)
void seg_sum_kernel(const float* __restrict__ data,
                    const int*   __restrict__ lengths,
                    const int*   __restrict__ offs,
                    float*       __restrict__ out) {
    __shared__ float lds[WAVES_PER_BLOCK * NBUF * SLOT_FLOATS];

    const int wave = threadIdx.x >> 5;
    const int lane = threadIdx.x & 31;
    const int seg  = blockIdx.x * WAVES_PER_BLOCK + wave;
    if (seg >= N_SEG) return;

    const int len   = lengths[seg];
    const int start = offs[seg];
    const int hi    = lane >> 4;    // 0: even row of pair, 1: odd row
    const int lx    = lane & 15;    // column group (4 floats)

    float* wbuf = &lds[wave * NBUF * SLOT_FLOATS];
    // Low 32 bits of a flat LDS-aperture address are the LDS byte offset.
    const uint32_t lbase = (uint32_t)(uintptr_t)wbuf + (uint32_t)(lane * 16);
    const uint8_t* gbase = (const uint8_t*)(data + (size_t)start * D) + lx * 16;

    if (len <= 0) {   // defensive (lengths are in [68,188])
        if (hi == 0) {
            float4 z = make_float4(0.f, 0.f, 0.f, 0.f);
            ((float4*)(out + (size_t)seg * D))[lx] = z;
        }
        return;
    }

    const int npairs = (len + 1) >> 1;

    // One async op = 512B = rows (2p, 2p+1); row index clamped so the
    // ASYNCcnt cadence stays constant through the tail.
    auto issue = [&](int pair, int slot) {
        int row = 2 * pair + hi;
        row = (row < len) ? row : (len - 1);
        uint64_t ga = (uint64_t)(uintptr_t)(gbase + (size_t)row * (D * 4));
        uint32_t la = lbase + (uint32_t)(slot * (SLOT_FLOATS * 4));
        asm volatile("global_load_async_to_lds_b128 %0, %1, off"
                     :: "v"(la), "v"(ga)
                     : "memory");
    };

    // Prologue: fill pipeline with PFD pair-ops.
    for (int j = 0; j < PFD; ++j) issue(j, j);

    float4 acc = make_float4(0.f, 0.f, 0.f, 0.f);
    const float4* lb = (const float4*)wbuf;   // slot*32 + lane float4 elements

    int i = 0;
    for (; i < npairs - 1; ++i) {
        // Pair i is the PFD-th oldest outstanding op -> wait <= PFD-1.
        asm volatile("s_wait_asynccnt 7" ::: "memory");
        float4 v = lb[(i & (NBUF - 1)) * 32 + lane];
        acc.x += v.x; acc.y += v.y; acc.z += v.z; acc.w += v.w;
        issue(i + PFD, (i + PFD) & (NBUF - 1));
    }

    // Peeled last pair: high half skips its add when len is odd (its row was
    // clamped to len-1, which the low half already owns).
    {
        asm volatile("s_wait_asynccnt 7" ::: "memory");
        float4 v = lb[(i & (NBUF - 1)) * 32 + lane];
        if (2 * i + hi < len) {
            acc.x += v.x; acc.y += v.y; acc.z += v.z; acc.w += v.w;
        }
    }
    // Leftover clamped prefetches drain at s_endpgm (implicit s_wait_idle).

    // Merge even-row partials (lanes 0-15) with odd-row partials (lanes 16-31).
    acc.x += __shfl_xor(acc.x, 16, 32);
    acc.y += __shfl_xor(acc.y, 16, 32);
    acc.z += __shfl_xor(acc.z, 16, 32);
    acc.w += __shfl_xor(acc.w, 16, 32);

    if (hi == 0) {
        ((float4*)(out + (size_t)seg * D))[lx] = acc;
    }
}

// ---------------------------------------------------------------------------
// Launcher
// ---------------------------------------------------------------------------
extern "C" void kernel_launch(void* const* d_in, const int* in_sizes, int n_in,
                              void* d_out, int out_size, void* d_ws, size_t ws_size,
                              hipStream_t stream) {
    const float* data    = (const float*)d_in[0];
    const int*   lengths = (const int*)d_in[1];
    float*       out     = (float*)d_out;
    int*         offs    = (int*)d_ws;          // 8192 ints of scratch

    seg_scan_kernel<<<1, SCAN_THREADS, 0, stream>>>(lengths, offs);
    seg_sum_kernel<<<N_SEG / WAVES_PER_BLOCK, WAVES_PER_BLOCK * 32, 0, stream>>>(
        data, lengths, offs, out);
}